// AugNODE_16509854286402
// MI455X (gfx1250) — compile-verified
//
#include <hip/hip_runtime.h>

typedef float v2f __attribute__((ext_vector_type(2)));
typedef float v8f __attribute__((ext_vector_type(8)));

#define BATCH    16384
#define VD       64      // VAR_DIM
#define HD       1024    // hidden dim
#define OUTD     32      // OUT_DIM
#define IND      32      // IN_DIM
#define MT       32      // batch rows per workgroup
#define NWAVES   8
#define NTHREADS (NWAVES * 32)
#define NSTEPS   8

#define WMMA_F32(A, B, C) \
    __builtin_amdgcn_wmma_f32_16x16x4_f32(false, (A), false, (B), (short)0, (C), false, false)

// Write one 16x16 C fragment (+time-column/bias term, optional ReLU) to LDS.
// C/D layout: lane covers column n; VGPR r holds row mbase + (lane>=16 ? 8:0) + r.
__device__ __forceinline__ void store_frag(float* __restrict__ actOut, int ldo, v8f c,
                                           int mbase, int n, float ext, bool relu,
                                           int half) {
    const int mofs = mbase + (half << 3);
#pragma unroll
    for (int r = 0; r < 8; ++r) {
        float v = c[r] + ext;
        if (relu) v = fmaxf(v, 0.0f);
        actOut[(mofs + r) * ldo + n] = v;
    }
}

// 2 M-tiles x 2 N-tiles per wave: each unique weight byte is loaded exactly
// once per workgroup per layer pass, and 4 independent WMMA accumulation
// chains hide the matrix-pipe latency.  N must be a multiple of 32.
__device__ __forceinline__ void layer_blocked(const float* __restrict__ W,
                                              const float* __restrict__ bias,
                                              const float* __restrict__ actIn, int K,
                                              float* __restrict__ actOut, int N,
                                              float t, bool relu, int wave, int lane) {
    const int wstride = K + 1;
    const int half    = lane >> 4;   // 0 or 1 -> which K pair of the chunk
    const int l16     = lane & 15;
    for (int np = wave; np < (N >> 5); np += NWAVES) {
        const int n0 = (np << 5) + l16;
        const int n1 = n0 + 16;
        v8f c00 = {}, c01 = {}, c10 = {}, c11 = {};
        const float* a0 = actIn + l16 * K + 2 * half;          // rows 0..15
        const float* a1 = actIn + (16 + l16) * K + 2 * half;   // rows 16..31
        const float* b0 = W + n0 * wstride + 2 * half;         // cols n0 group
        const float* b1 = W + n1 * wstride + 2 * half;         // cols n1 group
#pragma unroll 4
        for (int k = 0; k < K; k += 4) {
            v2f A0, A1, B0, B1;
            A0.x = a0[k]; A0.y = a0[k + 1];
            A1.x = a1[k]; A1.y = a1[k + 1];
            B0.x = b0[k]; B0.y = b0[k + 1];
            B1.x = b1[k]; B1.y = b1[k + 1];
            c00 = WMMA_F32(A0, B0, c00);
            c01 = WMMA_F32(A0, B1, c01);
            c10 = WMMA_F32(A1, B0, c10);
            c11 = WMMA_F32(A1, B1, c11);
        }
        const float e0 = t * W[n0 * wstride + K] + bias[n0];
        const float e1 = t * W[n1 * wstride + K] + bias[n1];
        store_frag(actOut, N, c00, 0,  n0, e0, relu, half);
        store_frag(actOut, N, c01, 0,  n1, e1, relu, half);
        store_frag(actOut, N, c10, 16, n0, e0, relu, half);
        store_frag(actOut, N, c11, 16, n1, e1, relu, half);
    }
}

// Narrow output layer (N == 64): one 16x16 tile per wave keeps all 8 waves busy.
__device__ __forceinline__ void layer_narrow(const float* __restrict__ W,
                                             const float* __restrict__ bias,
                                             const float* __restrict__ actIn, int K,
                                             float* __restrict__ actOut, int N,
                                             float t, bool relu, int wave, int lane) {
    const int wstride = K + 1;
    const int half    = lane >> 4;
    const int l16     = lane & 15;
    const int ntiles  = (MT / 16) * (N / 16);
    for (int tidx = wave; tidx < ntiles; tidx += NWAVES) {
        const int mt = (tidx & 1) << 4;
        const int n  = ((tidx >> 1) << 4) + l16;
        v8f c = {};
        const float* arow = actIn + (mt + l16) * K + 2 * half;
        const float* brow = W + n * wstride + 2 * half;
#pragma unroll 4
        for (int k = 0; k < K; k += 4) {
            v2f a, b;
            a.x = arow[k]; a.y = arow[k + 1];
            b.x = brow[k]; b.y = brow[k + 1];
            c = WMMA_F32(a, b, c);
        }
        const float e = t * W[n * wstride + K] + bias[n];
        store_frag(actOut, N, c, mt, n, e, relu, half);
    }
}

__launch_bounds__(NTHREADS, 1)
__global__ void augnode_rk4_kernel(
    const float* __restrict__ x, const float* __restrict__ aug,
    const float* __restrict__ W0, const float* __restrict__ b0,
    const float* __restrict__ W1, const float* __restrict__ b1,
    const float* __restrict__ W2, const float* __restrict__ b2,
    const float* __restrict__ W3, const float* __restrict__ b3,
    const float* __restrict__ W4, const float* __restrict__ b4,
    float* __restrict__ out) {
    // 288 KB of LDS: whole RK4 trajectory of a 32-row batch tile stays on-chip.
    __shared__ float actA[MT * HD];    // 128 KB ping
    __shared__ float actB[MT * HD];    // 128 KB pong
    __shared__ float ybase[MT * VD];   // y at step start
    __shared__ float ycur[MT * VD];    // stage input
    __shared__ float yacc[MT * VD];    // RK4 accumulator
    __shared__ float kbuf[MT * VD];    // odefunc output

    const int tid  = threadIdx.x;
    // Wave index as an SGPR: keeps tile loops scalar and EXEC all-ones
    // around every WMMA (hardware requirement).
    const int wave = __builtin_amdgcn_readfirstlane(tid >> 5);
    const int lane = tid & 31;
    const int row0 = blockIdx.x * MT;

    // y0 = concat(x, aug)
    for (int i = tid; i < MT * VD; i += NTHREADS) {
        const int r = i / VD, cidx = i % VD;
        const float v = (cidx < IND) ? x[(row0 + r) * IND + cidx]
                                     : aug[(row0 + r) * (VD - IND) + (cidx - IND)];
        ybase[i] = v;
        ycur[i]  = v;
    }
    __syncthreads();

    const float dt = 1.0f / (float)NSTEPS;

    for (int step = 0; step < NSTEPS; ++step) {
        const float t0 = (float)step * dt;
        for (int stage = 0; stage < 4; ++stage) {
            const float ts = (stage == 0) ? t0 : ((stage == 3) ? t0 + dt : t0 + 0.5f * dt);

            // ---- odefunc(ts, ycur) -> kbuf ----
            layer_blocked(W0, b0, ycur, VD, actA, HD, ts, true,  wave, lane);
            __syncthreads();
            layer_blocked(W1, b1, actA, HD, actB, HD, ts, true,  wave, lane);
            __syncthreads();
            layer_blocked(W2, b2, actB, HD, actA, HD, ts, true,  wave, lane);
            __syncthreads();
            layer_blocked(W3, b3, actA, HD, actB, HD, ts, true,  wave, lane);
            __syncthreads();
            layer_narrow (W4, b4, actB, HD, kbuf, VD, ts, false, wave, lane);
            __syncthreads();

            // ---- RK4 state update ----
            const float wk = (stage == 0 || stage == 3) ? dt / 6.0f : dt / 3.0f;
            const float cs = (stage < 2) ? 0.5f * dt : dt;  // next-stage offset
            for (int i = tid; i < MT * VD; i += NTHREADS) {
                const float k = kbuf[i];
                const float a = ((stage == 0) ? ybase[i] : yacc[i]) + wk * k;
                yacc[i] = a;
                if (stage < 3) ycur[i] = ybase[i] + cs * k;
            }
            __syncthreads();
        }
        // y_{n+1} = yacc
        for (int i = tid; i < MT * VD; i += NTHREADS) {
            const float v = yacc[i];
            ybase[i] = v;
            ycur[i]  = v;
        }
        __syncthreads();
    }

    // out = y1[:, :OUT_DIM]
    for (int i = tid; i < MT * OUTD; i += NTHREADS) {
        const int r = i / OUTD, cidx = i % OUTD;
        out[(row0 + r) * OUTD + cidx] = ybase[r * VD + cidx];
    }
}

extern "C" void kernel_launch(void* const* d_in, const int* in_sizes, int n_in,
                              void* d_out, int out_size, void* d_ws, size_t ws_size,
                              hipStream_t stream) {
    (void)in_sizes; (void)n_in; (void)d_ws; (void)ws_size; (void)out_size;
    const float* x   = (const float*)d_in[0];
    const float* aug = (const float*)d_in[1];
    const float* W0  = (const float*)d_in[2];
    const float* b0  = (const float*)d_in[3];
    const float* W1  = (const float*)d_in[4];
    const float* b1  = (const float*)d_in[5];
    const float* W2  = (const float*)d_in[6];
    const float* b2  = (const float*)d_in[7];
    const float* W3  = (const float*)d_in[8];
    const float* b3  = (const float*)d_in[9];
    const float* W4  = (const float*)d_in[10];
    const float* b4  = (const float*)d_in[11];
    float* out = (float*)d_out;

    dim3 grid(BATCH / MT);
    dim3 block(NTHREADS);
    augnode_rk4_kernel<<<grid, block, 0, stream>>>(x, aug, W0, b0, W1, b1, W2, b2,
                                                   W3, b3, W4, b4, out);
}